// CRNet_4054449128189
// MI455X (gfx1250) — compile-verified
//
#include <hip/hip_runtime.h>
#include <hip/hip_bf16.h>

// ---------------------------------------------------------------------------
// CRNet forward on MI455X (gfx1250).
// All 1x1 convs -> fp32 WMMA GEMM (v_wmma_f32_16x16x4_f32), BN/bias/ReLU/
// residual fused in epilogue.  kNN / softmax / gathers are VALU kernels.
// Transformer (N,k) tensors processed in 512-point chunks to bound workspace.
// GEMM has a guard-free, software-pipelined interior path (M%128==0, O%16==0,
// I%4==0 -- true for all heavy launches) and a guarded edge path for the tiny
// M=1 / O=3 / I=3 launches.
// ---------------------------------------------------------------------------

typedef float v2f __attribute__((ext_vector_type(2)));
typedef float v8f __attribute__((ext_vector_type(8)));

#define BN_INV 0.9999950000374997f   // 1/sqrt(1 + 1e-5)
#define NB 4                         // batch

// ============================ WMMA GEMM ====================================
// Y[b][o, m] = act( bn( W[o,:]*X[b][:,m] + bias[o] ) + R[b][o,m] )
// W element (o,i) at W[o*wO + i*wI];  X col-contig, row stride xR, batch xB.
// Y/R addressed as base + b*{y,r}B + o*{y,r}R + m*{y,r}C.
struct GemmArgs {
  const float* W; int wO; int wI;
  const float* X; long xB; int xR;
  const float* bias;
  const float* bnG; const float* bnB;
  const float* R; long rB; int rR; int rC;
  float* Y; long yB; int yR; int yC;
  int O; int I; int M; int act;     // act: 0=none, 1=relu
};

#define WMMA4(AF, B0, B1, B2, B3)                                                             \
  acc0 = __builtin_amdgcn_wmma_f32_16x16x4_f32(false, AF, false, B0, (short)0, acc0, false, false); \
  acc1 = __builtin_amdgcn_wmma_f32_16x16x4_f32(false, AF, false, B1, (short)0, acc1, false, false); \
  acc2 = __builtin_amdgcn_wmma_f32_16x16x4_f32(false, AF, false, B2, (short)0, acc2, false, false); \
  acc3 = __builtin_amdgcn_wmma_f32_16x16x4_f32(false, AF, false, B3, (short)0, acc3, false, false)

__global__ __launch_bounds__(256) void gemm_f32_wmma(GemmArgs a) {
  const int lane  = threadIdx.x & 31;
  const int wave  = threadIdx.x >> 5;
  const int oBase = blockIdx.y * 64 + (wave & 3) * 16;   // 4 waves along O
  const int mBase = blockIdx.x * 128 + (wave >> 2) * 64; // 2 waves along M
  const int l16   = lane & 15;
  const int kh    = (lane >> 4) * 2;                     // k offset 0 / 2
  const long b    = blockIdx.z;

  const float* Xb = a.X + b * a.xB;
  v8f acc0 = {}, acc1 = {}, acc2 = {}, acc3 = {};

  const int orow = oBase + l16;
  const int c0 = mBase + l16, c1 = c0 + 16, c2 = c0 + 32, c3 = c0 + 48;

  // Wave-uniform fast-path condition: whole 16x64 tile in range, K aligned.
  const bool interior = (oBase + 16 <= a.O) && (mBase + 64 <= a.M) && ((a.I & 3) == 0);

  if (interior) {
    const int  wi = a.wI, xr = a.xR;
    const long wStep = 4L * wi;
    const long xStep = 4L * xr;
    const float* wp = a.W + (long)orow * a.wO + (long)kh * wi;
    const float* xp = Xb + (long)kh * xr;

    // Prologue: fragments for k0 = 0.
    v2f af, b0, b1, b2, b3;
    af.x = wp[0];  af.y = wp[wi];
    b0.x = xp[c0]; b0.y = xp[xr + c0];
    b1.x = xp[c1]; b1.y = xp[xr + c1];
    b2.x = xp[c2]; b2.y = xp[xr + c2];
    b3.x = xp[c3]; b3.y = xp[xr + c3];

    // Software-pipelined main loop: load k+4 fragments, then WMMA on k.
    for (int k0 = 4; k0 < a.I; k0 += 4) {
      wp += wStep; xp += xStep;
      __builtin_prefetch(xp + 4 * xStep + c0, 0, 1);   // global_prefetch_b8
      v2f afn, b0n, b1n, b2n, b3n;
      afn.x = wp[0];  afn.y = wp[wi];
      b0n.x = xp[c0]; b0n.y = xp[xr + c0];
      b1n.x = xp[c1]; b1n.y = xp[xr + c1];
      b2n.x = xp[c2]; b2n.y = xp[xr + c2];
      b3n.x = xp[c3]; b3n.y = xp[xr + c3];
      WMMA4(af, b0, b1, b2, b3);
      af = afn; b0 = b0n; b1 = b1n; b2 = b2n; b3 = b3n;
    }
    WMMA4(af, b0, b1, b2, b3);
  } else {
    // Guarded edge path (M=1 global branch, O=3 regressor, I=3 pos_mlp).
    const bool oOK = orow < a.O;
    for (int k0 = 0; k0 < a.I; k0 += 4) {
      const int  ka = k0 + kh, kb = ka + 1;
      const bool kaOK = ka < a.I, kbOK = kb < a.I;

      v2f af; af.x = 0.0f; af.y = 0.0f;
      if (oOK) {
        const long wb = (long)orow * a.wO;
        if (kaOK) af.x = a.W[wb + (long)ka * a.wI];
        if (kbOK) af.y = a.W[wb + (long)kb * a.wI];
      }
      v2f b0; b0.x = 0.0f; b0.y = 0.0f;
      v2f b1 = b0, b2 = b0, b3 = b0;
      const long ra = (long)ka * a.xR, rb = (long)kb * a.xR;
      if (c0 < a.M) { if (kaOK) b0.x = Xb[ra + c0]; if (kbOK) b0.y = Xb[rb + c0]; }
      if (c1 < a.M) { if (kaOK) b1.x = Xb[ra + c1]; if (kbOK) b1.y = Xb[rb + c1]; }
      if (c2 < a.M) { if (kaOK) b2.x = Xb[ra + c2]; if (kbOK) b2.y = Xb[rb + c2]; }
      if (c3 < a.M) { if (kaOK) b3.x = Xb[ra + c3]; if (kbOK) b3.y = Xb[rb + c3]; }
      WMMA4(af, b0, b1, b2, b3);
    }
  }

  // Epilogue: VGPR r -> row r (lanes 0-15) / row r+8 (lanes 16-31); col = l16.
  float*       Yb = a.Y + b * a.yB;
  const float* Rb = a.R ? (a.R + b * a.rB) : nullptr;
  const int rofs = (lane >> 4) * 8;
  v8f accs[4] = {acc0, acc1, acc2, acc3};
  const int cols[4] = {c0, c1, c2, c3};
#pragma unroll
  for (int t = 0; t < 4; ++t) {
    const int col = cols[t];
    if (col >= a.M) continue;
#pragma unroll
    for (int rr = 0; rr < 8; ++rr) {
      const int oo = oBase + rofs + rr;
      if (oo < a.O) {
        float v = accs[t][rr];
        if (a.bias) v += a.bias[oo];
        if (a.bnG)  v = v * (a.bnG[oo] * BN_INV) + a.bnB[oo];
        if (Rb)     v += Rb[(long)oo * a.rR + (long)col * a.rC];
        if (a.act == 1) v = fmaxf(v, 0.0f);
        Yb[(long)oo * a.yR + (long)col * a.yC] = v;
      }
    }
  }
}

// ============================ kNN top-16 ===================================
// pts (B,3,N); idx (B,N,16).  d = |n|^2 + |m|^2 - 2 n.m (reference formula).
__global__ __launch_bounds__(128) void knn_kernel(const float* pts, int* idx, int N) {
  const int b = blockIdx.y;
  const int n = blockIdx.x * 128 + threadIdx.x;
  const float* px = pts + (long)b * 3 * N;
  const float* py = px + N;
  const float* pz = py + N;

  float xn = px[n], yn = py[n], zn = pz[n];
  float sqn = xn * xn + yn * yn + zn * zn;

  float bd[16]; int bi[16];
#pragma unroll
  for (int j = 0; j < 16; ++j) { bd[j] = __builtin_inff(); bi[j] = 0; }

  __shared__ float sx[128], sy[128], sz[128], sq[128];
  for (int t0 = 0; t0 < N; t0 += 128) {
    __syncthreads();
    const int m = t0 + threadIdx.x;           // N is a multiple of 128
    float mx = px[m], my = py[m], mz = pz[m];
    sx[threadIdx.x] = mx; sy[threadIdx.x] = my; sz[threadIdx.x] = mz;
    sq[threadIdx.x] = mx * mx + my * my + mz * mz;
    __syncthreads();
    for (int u = 0; u < 128; ++u) {
      const float d = sqn + sq[u] - 2.0f * (xn * sx[u] + yn * sy[u] + zn * sz[u]);
      if (d < bd[15]) {
        float dd = d; int ii = t0 + u;
#pragma unroll
        for (int j = 0; j < 16; ++j) {
          if (dd < bd[j]) {
            float td = bd[j]; int ti = bi[j];
            bd[j] = dd; bi[j] = ii; dd = td; ii = ti;
          }
        }
      }
    }
  }
#pragma unroll
  for (int j = 0; j < 16; ++j) idx[((long)b * N + n) * 16 + j] = bi[j];
}

// ========================= small elementwise kernels =======================
__global__ void ew_add_snr(const float* f, const float* snr, float* o, long CN, long tot) {
  long i = blockIdx.x * 256L + threadIdx.x;
  if (i < tot) o[i] = f[i] + snr[i / CN];
}

// op: 0 = mean, 1 = max.  One block per (b*C + c) row of length N.
__global__ __launch_bounds__(128) void colreduce(const float* X, float* out, int N, int op) {
  const float* row = X + (long)blockIdx.x * N;
  float acc = op ? -__builtin_inff() : 0.0f;
  for (int i = threadIdx.x; i < N; i += 128) {
    float v = row[i];
    acc = op ? fmaxf(acc, v) : acc + v;
  }
  __shared__ float s[128];
  s[threadIdx.x] = acc; __syncthreads();
  for (int st = 64; st > 0; st >>= 1) {
    if ((int)threadIdx.x < st)
      s[threadIdx.x] = op ? fmaxf(s[threadIdx.x], s[threadIdx.x + st])
                          : s[threadIdx.x] + s[threadIdx.x + st];
    __syncthreads();
  }
  if (threadIdx.x == 0) out[blockIdx.x] = op ? s[0] : s[0] / (float)N;
}

__global__ void ew_gate(const float* feat, const float* loc, const float* glob,
                        float* o, int N, long tot) {
  long i = blockIdx.x * 256L + threadIdx.x;
  if (i >= tot) return;
  long bc = i / N;
  float x = loc[i] + glob[bc];
  o[i] = feat[i] * (1.0f / (1.0f + expf(-x)));
}

__global__ void ew_concat(const float* outf, const float* gmax, float* f2, int N, long tot) {
  long i = blockIdx.x * 256L + threadIdx.x;
  if (i >= tot) return;
  long n = i % N; long c = (i / N) % 640; long b = i / (640L * N);
  f2[i] = (c < 320) ? outf[(b * 320 + c) * N + n] : gmax[b * 320 + (c - 320)];
}

__global__ void ew_posrel(const float* pts, const int* idx, float* PR, int N, int n0, long tot) {
  long i = blockIdx.x * 256L + threadIdx.x;
  if (i >= tot) return;
  int j = (int)(i & 15); long q = i >> 4;
  int nn = (int)(q % 512); q /= 512;
  int c = (int)(q % 3); int b = (int)(q / 3);
  int n = n0 + nn;
  int id = idx[((long)b * N + n) * 16 + j];
  const float* p = pts + ((long)b * 3 + c) * N;
  PR[i] = p[n] - p[id];
}

__global__ void ew_qkpe(const float* qry, const float* key, const int* idx,
                        const float* PE, float* QK, int N, int n0, long tot) {
  long i = blockIdx.x * 256L + threadIdx.x;
  if (i >= tot) return;
  int j = (int)(i & 15); long q = i >> 4;
  int nn = (int)(q % 512); q /= 512;
  int c = (int)(q % 128); int b = (int)(q / 128);
  int n = n0 + nn;
  int id = idx[((long)b * N + n) * 16 + j];
  long ch = ((long)b * 128 + c) * N;
  QK[i] = qry[ch + n] - key[ch + id] + PE[i];
}

__global__ void ew_softmax_agg(const float* AT, const float* PE, const float* val,
                               float* AGG, int N, int n0, long tot) {
  long i = blockIdx.x * 256L + threadIdx.x;
  if (i >= tot) return;
  int nn = (int)(i % 512); long q = i / 512;
  int c = (int)(q % 128); int b = (int)(q / 128);
  long base = i * 16L;
  float v[16]; float m = -__builtin_inff();
#pragma unroll
  for (int j = 0; j < 16; ++j) { v[j] = AT[base + j]; m = fmaxf(m, v[j]); }
  float s = 0.0f;
#pragma unroll
  for (int j = 0; j < 16; ++j) { v[j] = expf(v[j] - m); s += v[j]; }
  const float inv = 1.0f / s;
  long ch = ((long)b * 128 + c) * N + n0 + nn;
  float vv = val[ch];
  float acc = 0.0f;
#pragma unroll
  for (int j = 0; j < 16; ++j) acc += v[j] * inv * (vv + PE[base + j]);
  AGG[ch] = acc;
}

__global__ void ew_dup(const float* pf, float* cat, int N, int r, long tot) {
  long i = blockIdx.x * 256L + threadIdx.x;
  if (i >= tot) return;
  long NR = (long)N * r;
  long m = i % NR; long q = i / NR;
  int c = (int)(q % 320); int b = (int)(q / 320);
  cat[((long)b * 640 + 320 + c) * NR + m] = pf[((long)b * 320 + c) * N + m / r];
}

__global__ void ew_finalize(const float* pts, const float* off, float* out,
                            int N, int r, int tr, long tot) {
  long i = blockIdx.x * 256L + threadIdx.x;
  if (i >= tot) return;
  long NR = (long)N * r;
  long m = i % NR; long q = i / NR;
  int c = (int)(q % 3); int b = (int)(q / 3);
  float v = pts[((long)b * 3 + c) * N + m / r] + tanhf(off[((long)b * 3 + c) * NR + m]);
  if (tr) out[((long)b * NR + m) * 3 + c] = v;
  else    out[((long)b * 3 + c) * NR + m] = v;
}

// ============================ host-side plumbing ===========================
struct Conv { const float* b; const float* w; };   // jax leaf order: 'b' < 'w'
struct BNp  { const float* b; const float* g; };   // 'b' < 'g'

struct SubP {
  Conv dcv;                              // 'deconv' conv(320,320)
  Conv m2a, m2b;                         // extractor.mlp2
  Conv am0; BNp am1; Conv am2;           // tf.attn_mlp
  Conv endc, keyc;                       // tf.end, tf.key
  Conv pm0; BNp pm1; Conv pm2;           // tf.pos_mlp
  Conv qryc, startc, valc;               // tf.query, tf.start, tf.value
  Conv g0; BNp g1; Conv g2; BNp g3;      // global
  Conv l0; BNp l1; Conv l2; BNp l3;      // local
  Conv r0, r1;                           // regressor
  const float* dw;                       // up.deconv_w (64,320,r)
  Conv u0, u1;                           // up.mlp1
  Conv c1, c2, sc;                       // up.mlp_res
};

static SubP parse_sub(void* const* din, int& pi) {
  auto F = [&]() { return (const float*)din[pi++]; };
  SubP s;
  s.dcv = {F(), F()};
  s.m2a = {F(), F()}; s.m2b = {F(), F()};
  s.am0 = {F(), F()}; s.am1 = {F(), F()}; s.am2 = {F(), F()};
  s.endc = {F(), F()};
  s.keyc = {F(), F()};
  s.pm0 = {F(), F()}; s.pm1 = {F(), F()}; s.pm2 = {F(), F()};
  s.qryc = {F(), F()};
  s.startc = {F(), F()};
  s.valc = {F(), F()};
  s.g0 = {F(), F()}; s.g1 = {F(), F()}; s.g2 = {F(), F()}; s.g3 = {F(), F()};
  s.l0 = {F(), F()}; s.l1 = {F(), F()}; s.l2 = {F(), F()}; s.l3 = {F(), F()};
  s.r0 = {F(), F()}; s.r1 = {F(), F()};
  s.dw = F();
  s.u0 = {F(), F()}; s.u1 = {F(), F()};
  s.c1 = {F(), F()}; s.c2 = {F(), F()}; s.sc = {F(), F()};
  return s;
}

struct WS {
  float *fs, *lh, *lo, *gmean, *gh, *gout, *outf, *gmax, *f2, *m2h, *f3;
  float *x128, *keyb, *qryb, *valb, *agg, *tfout, *pfeat, *uph, *h64;
  float *cat, *t1, *t3, *regh, *offv, *ufA, *ufB, *ptsA, *ptsB;
  float *PR, *PEH, *PE, *QK, *AH, *AT;
  int* idx;
};

static WS make_ws(void* ws) {
  float* p = (float*)ws;
  auto A = [&](long n) { float* r = p; p += n; return r; };
  const long NM = 4096;             // max N (s3), max N*r (s2/s3)
  const long MK = 8192;             // per-batch chunk cols = 512*16
  WS w;
  w.fs    = A(NB * 320L * NM);  w.lh   = A(NB * 128L * NM);
  w.lo    = A(NB * 320L * NM);  w.gmean= A(NB * 320L);
  w.gh    = A(NB * 128L);       w.gout = A(NB * 320L);
  w.outf  = A(NB * 320L * NM);  w.gmax = A(NB * 320L);
  w.f2    = A(NB * 640L * NM);  w.m2h  = A(NB * 640L * NM);
  w.f3    = A(NB * 320L * NM);
  w.x128  = A(NB * 128L * NM);  w.keyb = A(NB * 128L * NM);
  w.qryb  = A(NB * 128L * NM);  w.valb = A(NB * 128L * NM);
  w.agg   = A(NB * 128L * NM);  w.tfout= A(NB * 320L * NM);
  w.pfeat = A(NB * 320L * NM);  w.uph  = A(NB * 128L * NM);
  w.h64   = A(NB * 64L * NM);
  w.cat   = A(NB * 640L * NM);  w.t1   = A(NB * 320L * NM);
  w.t3    = A(NB * 320L * NM);  w.regh = A(NB * 128L * NM);
  w.offv  = A(NB * 3L * NM);
  w.ufA   = A(NB * 320L * NM);  w.ufB  = A(NB * 320L * NM);
  w.ptsA  = A(NB * 3L * NM);    w.ptsB = A(NB * 3L * NM);
  w.PR    = A(NB * 3L * MK);    w.PEH  = A(NB * 64L * MK);
  w.PE    = A(NB * 128L * MK);  w.QK   = A(NB * 128L * MK);
  w.AH    = A(NB * 512L * MK);  w.AT   = A(NB * 128L * MK);
  w.idx   = (int*)A(NB * NM * 16L);
  return w;
}

static void gemm(hipStream_t st, const float* W, int wO, int wI,
                 const float* X, long xB, int xR,
                 const float* bias, const float* bnG, const float* bnB,
                 const float* R, long rB, int rR, int rC,
                 float* Y, long yB, int yR, int yC,
                 int O, int I, int M, int act) {
  GemmArgs a;
  a.W = W; a.wO = wO; a.wI = wI;
  a.X = X; a.xB = xB; a.xR = xR;
  a.bias = bias; a.bnG = bnG; a.bnB = bnB;
  a.R = R; a.rB = rB; a.rR = rR; a.rC = rC;
  a.Y = Y; a.yB = yB; a.yR = yR; a.yC = yC;
  a.O = O; a.I = I; a.M = M; a.act = act;
  dim3 g((unsigned)((M + 127) / 128), (unsigned)((O + 63) / 64), NB);
  gemm_f32_wmma<<<g, dim3(256), 0, st>>>(a);
}

// Standard 1x1 conv over (B,C,N) tensors (Y/R shaped (B,O,N)).
static void conv1x1(hipStream_t st, Conv c, int O, int I, const float* X, int N, float* Y,
                    const float* bnG = nullptr, const float* bnB = nullptr, int act = 0,
                    const float* R = nullptr) {
  gemm(st, c.w, I, 1, X, (long)I * N, N, c.b, bnG, bnB,
       R, (long)O * N, N, 1, Y, (long)O * N, N, 1, O, I, N, act);
}

// Conv over chunked (B,C,512,16) tensors.
static void convchunk(hipStream_t st, Conv c, int O, int I, const float* X, float* Y,
                      const float* bnG, const float* bnB, int act) {
  const int M = 512 * 16;
  gemm(st, c.w, I, 1, X, (long)I * M, M, c.b, bnG, bnB,
       nullptr, 0, 0, 0, Y, (long)O * M, M, 1, O, I, M, act);
}

#define EW(K, TOT, ...) K<<<dim3((unsigned)(((TOT) + 255) / 256)), dim3(256), 0, st>>>(__VA_ARGS__)

static void run_subnetwork(hipStream_t st, const SubP& P, WS& w,
                           const float* pts, const float* feat, const float* snr,
                           int N, int r, float* ptsOut, float* featOut, float* finalOut) {
  const long NR = (long)N * r;

  // fs = feat + snr[b]
  EW(ew_add_snr, NB * 320L * N, feat, snr, w.fs, 320L * N, NB * 320L * N);
  // local att_branch
  conv1x1(st, P.l0, 128, 320, w.fs, N, w.lh, P.l1.g, P.l1.b, 1);
  conv1x1(st, P.l2, 320, 128, w.lh, N, w.lo, P.l3.g, P.l3.b, 0);
  // global att_branch on mean
  colreduce<<<dim3(NB * 320), dim3(128), 0, st>>>(w.fs, w.gmean, N, 0);
  gemm(st, P.g0.w, 320, 1, w.gmean, 320, 1, P.g0.b, P.g1.g, P.g1.b,
       nullptr, 0, 0, 0, w.gh, 128, 1, 1, 128, 320, 1, 1);
  gemm(st, P.g2.w, 128, 1, w.gh, 128, 1, P.g2.b, P.g3.g, P.g3.b,
       nullptr, 0, 0, 0, w.gout, 320, 1, 1, 320, 128, 1, 0);
  // out_feature = feat * sigmoid(local + global)
  EW(ew_gate, NB * 320L * N, feat, w.lo, w.gout, w.outf, N, NB * 320L * N);
  // extractor: concat(out_feature, max) -> mlp2
  colreduce<<<dim3(NB * 320), dim3(128), 0, st>>>(w.outf, w.gmax, N, 1);
  EW(ew_concat, NB * 640L * N, w.outf, w.gmax, w.f2, N, NB * 640L * N);
  conv1x1(st, P.m2a, 640, 640, w.f2, N, w.m2h, nullptr, nullptr, 1);
  conv1x1(st, P.m2b, 320, 640, w.m2h, N, w.f3, nullptr, nullptr, 0);
  // transformer
  conv1x1(st, P.startc, 128, 320, w.f3, N, w.x128);
  conv1x1(st, P.keyc, 128, 128, w.x128, N, w.keyb);
  conv1x1(st, P.qryc, 128, 128, w.x128, N, w.qryb);
  conv1x1(st, P.valc, 128, 128, w.x128, N, w.valb);
  knn_kernel<<<dim3(N / 128, NB), dim3(128), 0, st>>>(pts, w.idx, N);
  for (int n0 = 0; n0 < N; n0 += 512) {
    EW(ew_posrel, NB * 3L * 512 * 16, pts, w.idx, w.PR, N, n0, NB * 3L * 512 * 16);
    convchunk(st, P.pm0, 64, 3, w.PR, w.PEH, P.pm1.g, P.pm1.b, 1);
    convchunk(st, P.pm2, 128, 64, w.PEH, w.PE, nullptr, nullptr, 0);
    EW(ew_qkpe, NB * 128L * 512 * 16, w.qryb, w.keyb, w.idx, w.PE, w.QK, N, n0,
       NB * 128L * 512 * 16);
    convchunk(st, P.am0, 512, 128, w.QK, w.AH, P.am1.g, P.am1.b, 1);
    convchunk(st, P.am2, 128, 512, w.AH, w.AT, nullptr, nullptr, 0);
    EW(ew_softmax_agg, NB * 128L * 512, w.AT, w.PE, w.valb, w.agg, N, n0, NB * 128L * 512);
  }
  conv1x1(st, P.endc, 320, 128, w.agg, N, w.tfout, nullptr, nullptr, 0, w.f3);
  // deconv conv + up_unit
  conv1x1(st, P.dcv, 320, 320, w.tfout, N, w.pfeat);
  conv1x1(st, P.u0, 128, 320, w.pfeat, N, w.uph, nullptr, nullptr, 1);
  conv1x1(st, P.u1, 64, 128, w.uph, N, w.h64);
  for (int t = 0; t < r; ++t) {   // ConvTranspose1d as r strided GEMMs -> cat rows 0..319
    gemm(st, P.dw + t, r, 320 * r, w.h64, 64L * N, N, nullptr, nullptr, nullptr,
         nullptr, 0, 0, 0, w.cat + t, 640L * NR, (int)NR, r, 320, 64, N, 0);
  }
  EW(ew_dup, NB * 320L * NR, w.pfeat, w.cat, N, r, NB * 320L * NR);
  // mlp_res: relu(c2(relu(c1(cat))) + sc(cat))
  conv1x1(st, P.c1, 320, 640, w.cat, (int)NR, w.t1, nullptr, nullptr, 1);
  conv1x1(st, P.sc, 320, 640, w.cat, (int)NR, w.t3);
  conv1x1(st, P.c2, 320, 320, w.t1, (int)NR, featOut, nullptr, nullptr, 1, w.t3);
  // regressor + tanh offset
  conv1x1(st, P.r0, 128, 320, featOut, (int)NR, w.regh, nullptr, nullptr, 1);
  conv1x1(st, P.r1, 3, 128, w.regh, (int)NR, w.offv);
  EW(ew_finalize, NB * 3L * NR, pts, w.offv, finalOut ? finalOut : ptsOut,
     N, r, finalOut ? 1 : 0, NB * 3L * NR);
}

extern "C" void kernel_launch(void* const* d_in, const int* in_sizes, int n_in,
                              void* d_out, int out_size, void* d_ws, size_t ws_size,
                              hipStream_t stream) {
  (void)in_sizes; (void)n_in; (void)out_size; (void)ws_size;
  const float* pts0 = (const float*)d_in[0];   // (B,3,1024)
  const float* feat0 = (const float*)d_in[1];  // (B,320,1024)
  const float* snr = (const float*)d_in[2];    // (B,)
  int pi = 3;
  SubP S1 = parse_sub(d_in, pi);
  SubP S2 = parse_sub(d_in, pi);
  SubP S3 = parse_sub(d_in, pi);
  WS w = make_ws(d_ws);

  run_subnetwork(stream, S1, w, pts0, feat0, snr, 1024, 2, w.ptsA, w.ufA, nullptr);
  run_subnetwork(stream, S2, w, w.ptsA, w.ufA, snr, 2048, 2, w.ptsB, w.ufB, nullptr);
  run_subnetwork(stream, S3, w, w.ptsB, w.ufB, snr, 4096, 1, w.ptsA, w.ufA, (float*)d_out);
}